// EigenLoRA_A_26963804685044
// MI455X (gfx1250) — compile-verified
//
#include <hip/hip_runtime.h>

typedef __attribute__((ext_vector_type(2))) float v2f;
typedef __attribute__((ext_vector_type(4))) float v4f;
typedef __attribute__((ext_vector_type(8))) float v8f;

#define N_TOKENS 16384
#define IN_F     4096
#define NCOMP    64
#define OUT_F    64

#define CK        64            // k-chunk staged in LDS per buffer
#define NCHUNK    (IN_F / CK)   // 64
#define JGS       (CK / 8)      // 8 eight-k groups per chunk
#define LDSTRIDE  68            // dwords/row: 272B = 16B aligned; banks 4*o mod 64 -> conflict-free b128
#define BUFDW     (OUT_F * LDSTRIDE)   // dwords per LDS buffer (4352)
#define NBUF      3             // triple buffer, DMA distance 2
#define XPF       8             // x prefetch depth in 8-k groups

// ---------------------------------------------------------------------------
// Kernel 1: reconsT[o][i] = sum_c components[i][c] * loadings[c][o]
// ---------------------------------------------------------------------------
__global__ __launch_bounds__(256) void eigenlora_recons(
    const float* __restrict__ comp, const float* __restrict__ loadings,
    float* __restrict__ recT) {
  const int i = blockIdx.x * 256 + threadIdx.x;   // 0..4095
  float cr[NCOMP];
#pragma unroll
  for (int c4 = 0; c4 < NCOMP / 4; ++c4) {
    v4f v = *(const v4f*)(comp + (size_t)i * NCOMP + c4 * 4);
    cr[c4 * 4 + 0] = v.x; cr[c4 * 4 + 1] = v.y;
    cr[c4 * 4 + 2] = v.z; cr[c4 * 4 + 3] = v.w;
  }
  for (int o = 0; o < OUT_F; ++o) {
    float acc = 0.f;
#pragma unroll
    for (int c = 0; c < NCOMP; ++c)
      acc = __builtin_fmaf(cr[c], loadings[c * OUT_F + o], acc);
    recT[(size_t)o * IN_F + i] = acc;
  }
}

// ---------------------------------------------------------------------------
// CDNA5 async global->LDS staging (ASYNCcnt-tracked DMA, GVS addressing)
// ---------------------------------------------------------------------------
__device__ __forceinline__ void stage_chunk_async(unsigned ldsbase_byte,
                                                  const float* recT,
                                                  int chunk, int tid) {
#pragma unroll
  for (int it = 0; it < 4; ++it) {
    const int f  = tid + 256 * it;         // float4 index 0..1023
    const int o  = f >> 4;                 // 16 float4 per 64-float row
    const int c4 = f & 15;
    unsigned ldsb = ldsbase_byte + (unsigned)((o * LDSTRIDE + c4 * 4) * 4);
    unsigned voff = (unsigned)((o * IN_F + chunk * CK + c4 * 4) * 4);
    asm volatile("global_load_async_to_lds_b128 %0, %1, %2 offset:0"
                 :: "v"(ldsb), "v"(voff), "s"(recT) : "memory");
  }
}
__device__ __forceinline__ void wait_async_le4() {
  asm volatile("s_wait_asynccnt 0x4" ::: "memory");
}
__device__ __forceinline__ void wait_async_le0() {
  asm volatile("s_wait_asynccnt 0x0" ::: "memory");
}

#define WMMA_F32(A, B, C) __builtin_amdgcn_wmma_f32_16x16x4_f32( \
    false, (A), false, (B), (short)0, (C), false, false)

// ---------------------------------------------------------------------------
// One chunk: 8 jg-groups; per group 1 x-load (depth-8 ring), 4 LDS B-frag
// loads (prefetched one group ahead), 8 WMMA.
// ---------------------------------------------------------------------------
template <bool XPREFETCH>
__device__ __forceinline__ void compute_chunk(
    const float* __restrict__ ldsbuf, const float* __restrict__ xlane,
    int gbase, v4f (&axr)[XPF], v8f (&acc)[4], int lane16, int laneHi) {
  const int boff = 4 * laneHi;             // dword offset within row
  v4f b0 = *(const v4f*)(ldsbuf + (0 * 16 + lane16) * LDSTRIDE + boff);
  v4f b1 = *(const v4f*)(ldsbuf + (1 * 16 + lane16) * LDSTRIDE + boff);
  v4f b2 = *(const v4f*)(ldsbuf + (2 * 16 + lane16) * LDSTRIDE + boff);
  v4f b3 = *(const v4f*)(ldsbuf + (3 * 16 + lane16) * LDSTRIDE + boff);
#pragma unroll
  for (int jg = 0; jg < JGS; ++jg) {
    const v4f a4 = axr[jg];                // consume ring slot
    if (XPREFETCH)                         // refill 8 groups ahead
      axr[jg] = __builtin_nontemporal_load(
          (const v4f*)(xlane + (size_t)(gbase + jg + XPF) * 8));

    const v4f c0 = b0, c1 = b1, c2 = b2, c3 = b3;
    if (jg < JGS - 1) {                    // prefetch next group's B frags
      const int nb = (jg + 1) * 8 + boff;
      b0 = *(const v4f*)(ldsbuf + (0 * 16 + lane16) * LDSTRIDE + nb);
      b1 = *(const v4f*)(ldsbuf + (1 * 16 + lane16) * LDSTRIDE + nb);
      b2 = *(const v4f*)(ldsbuf + (2 * 16 + lane16) * LDSTRIDE + nb);
      b3 = *(const v4f*)(ldsbuf + (3 * 16 + lane16) * LDSTRIDE + nb);
    }

    v2f aLo; aLo.x = a4.x; aLo.y = a4.y;   // k-slots (8j, 8j+1, 8j+4, 8j+5)
    v2f aHi; aHi.x = a4.z; aHi.y = a4.w;   // k-slots (+2)
    v2f b;
    b.x = c0.x; b.y = c0.y; acc[0] = WMMA_F32(aLo, b, acc[0]);
    b.x = c1.x; b.y = c1.y; acc[1] = WMMA_F32(aLo, b, acc[1]);
    b.x = c2.x; b.y = c2.y; acc[2] = WMMA_F32(aLo, b, acc[2]);
    b.x = c3.x; b.y = c3.y; acc[3] = WMMA_F32(aLo, b, acc[3]);
    b.x = c0.z; b.y = c0.w; acc[0] = WMMA_F32(aHi, b, acc[0]);
    b.x = c1.z; b.y = c1.w; acc[1] = WMMA_F32(aHi, b, acc[1]);
    b.x = c2.z; b.y = c2.w; acc[2] = WMMA_F32(aHi, b, acc[2]);
    b.x = c3.z; b.y = c3.w; acc[3] = WMMA_F32(aHi, b, acc[3]);
  }
}

// ---------------------------------------------------------------------------
// Kernel 2: out = x @ recons. Triple-buffered async LDS staging, one barrier
// per chunk, single compute instantiation in the steady-state loop.
// ---------------------------------------------------------------------------
__global__ __launch_bounds__(256) void eigenlora_gemm(
    const float* __restrict__ x, const float* __restrict__ recT,
    float* __restrict__ out) {
  __shared__ float ldsB[NBUF * BUFDW];     // 3 x 17408B buffers

  const int tid    = threadIdx.x;
  const int lane   = tid & 31;
  const int wave   = tid >> 5;
  const int lane16 = lane & 15;
  const int laneHi = lane >> 4;
  const int m0     = blockIdx.x * 128 + wave * 16;

  const unsigned ldsbase0 = (unsigned)(unsigned long long)(&ldsB[0]);
  const unsigned bufbytes = BUFDW * 4;

  v8f acc[4] = {};
  // lane L<16: cols 8g..8g+3 of row m0+L; lane L+16: cols 8g+4..8g+7
  const float* xlane = x + (size_t)(m0 + lane16) * IN_F + 4 * laneHi;

  // ---- prologue: fill x ring (g = 0..7), kick DMA for chunks 0 and 1 ----
  v4f axr[XPF];
#pragma unroll
  for (int p = 0; p < XPF; ++p)
    axr[p] = __builtin_nontemporal_load((const v4f*)(xlane + (size_t)p * 8));
  stage_chunk_async(ldsbase0 + 0 * bufbytes, recT, 0, tid);
  stage_chunk_async(ldsbase0 + 1 * bufbytes, recT, 1, tid);

  int cur = 0;                             // buffer holding chunk c
  int nx2 = 2;                             // buffer for chunk c+2
#pragma unroll 1
  for (int c = 0; c < NCHUNK - 1; ++c) {
    wait_async_le4();                      // own chunk-c DMAs landed
    __syncthreads();                       // chunk c visible to all waves;
                                           // all waves done reading buf nx2
    if (c < NCHUNK - 2)
      stage_chunk_async(ldsbase0 + (unsigned)nx2 * bufbytes, recT, c + 2, tid);

    compute_chunk<true>(ldsB + cur * BUFDW, xlane, c * JGS,
                        axr, acc, lane16, laneHi);

    cur = (cur == NBUF - 1) ? 0 : cur + 1;
    nx2 = (nx2 == NBUF - 1) ? 0 : nx2 + 1;
  }
  // ---- last chunk (no x prefetch, no further DMA) ----
  wait_async_le0();
  __syncthreads();
  compute_chunk<false>(ldsB + cur * BUFDW, xlane, (NCHUNK - 1) * JGS,
                       axr, acc, lane16, laneHi);

  // C/D layout: VGPR r -> row m0 + r + 8*laneHi, col lane16 (+16*t)
#pragma unroll
  for (int r = 0; r < 8; ++r) {
    const int m = m0 + r + 8 * laneHi;
#pragma unroll
    for (int t = 0; t < 4; ++t)
      __builtin_nontemporal_store(acc[t][r],
          out + (size_t)m * OUT_F + t * 16 + lane16);
  }
}

// ---------------------------------------------------------------------------
extern "C" void kernel_launch(void* const* d_in, const int* in_sizes, int n_in,
                              void* d_out, int out_size, void* d_ws, size_t ws_size,
                              hipStream_t stream) {
  (void)in_sizes; (void)n_in; (void)out_size; (void)ws_size;
  const float* x        = (const float*)d_in[0];
  const float* comp     = (const float*)d_in[1];
  const float* loadings = (const float*)d_in[2];
  float* out  = (float*)d_out;
  float* recT = (float*)d_ws;              // 64*4096*4 = 1 MB scratch

  hipLaunchKernelGGL(eigenlora_recons, dim3(IN_F / 256), dim3(256), 0, stream,
                     comp, loadings, recT);
  hipLaunchKernelGGL(eigenlora_gemm, dim3(N_TOKENS / 128), dim3(256), 0, stream,
                     x, recT, out);
}